// ProjectLayerJrn_68109591380310
// MI455X (gfx1250) — compile-verified
//
#include <hip/hip_runtime.h>

// ---------------- constants from the reference ----------------
#define N_CAMS   5
#define N_JOINTS 15
#define HMH      128
#define HMW      240
#define HW       (HMH * HMW)          // 30720
#define CUBE_X   80
#define CUBE_Y   80
#define CUBE_Z   20
#define NBINS    (CUBE_X * CUBE_Y * CUBE_Z)   // 128000

typedef __attribute__((ext_vector_type(2))) float v2f;
typedef __attribute__((ext_vector_type(8))) float v8f;

// grid point for linear voxel index n (meshgrid 'ij', ravel order ix,iy,iz)
__device__ __forceinline__ void grid_point(int n, float& gx, float& gy, float& gz) {
    int iz = n % CUBE_Z;
    int t  = n / CUBE_Z;
    int iy = t % CUBE_Y;
    int ix = t / CUBE_Y;
    gx = -4000.0f + (float)ix * (8000.0f / 79.0f);            // + center 0
    gy = -4000.0f + (float)iy * (8000.0f / 79.0f) - 500.0f;   // + center -500
    gz = -1000.0f + (float)iz * (2000.0f / 19.0f) + 800.0f;   // + center 800
}

struct Proj {
    float xy2x, xy2y;   // heatmap-space coords BEFORE sg clipping (the xys output)
    float sgx, sgy;     // normalized sample coords, clipped to [-1.1, 1.1]
    float bound;        // 0/1 in-image mask (pre-clip)
};

// distortion + crop/affine transform (per_cam body after xcam), faithful to reference
__device__ __forceinline__ Proj cam_transform(
    float xcx, float xcy, float xcz,
    float k0, float k1, float k2, float p0, float p1,
    float f0, float f1, float c0, float c1,
    float ctr0, float ctr1, float scl0)
{
    float z   = xcz + 1e-5f;
    float y0  = xcx / z;
    float y1  = xcy / z;
    float r2  = y0 * y0 + y1 * y1;
    float radial = 1.0f + k0 * r2 + k1 * (r2 * r2) + k2 * (r2 * r2 * r2);
    float tan_   = p0 * y1 + p1 * y0;
    float s      = radial + tan_;
    float yy0 = y0 * s + p1 * r2;
    float yy1 = y1 * s + p0 * r2;
    float px = f0 * yy0 + c0;
    float py = f1 * yy1 + c1;

    float width  = ctr0 * 2.0f;
    float height = ctr1 * 2.0f;
    float b = (px >= 0.0f && py >= 0.0f && px < width && py < height) ? 1.0f : 0.0f;
    float mx = fmaxf(width, height);
    px = fminf(fmaxf(px, -1.0f), mx);
    py = fminf(fmaxf(py, -1.0f), mx);

    float hsc = 200.0f * scl0;
    float xt = px * (960.0f / hsc) + 960.0f * (-ctr0 / hsc + 0.5f);
    float yt = py * (512.0f / hsc) + 512.0f * (-ctr1 / hsc + 0.5f);

    Proj r;
    r.xy2x = xt * (240.0f / 960.0f);
    r.xy2y = yt * (128.0f / 512.0f);
    r.sgx  = fminf(fmaxf(r.xy2x / 239.0f * 2.0f - 1.0f, -1.1f), 1.1f);
    r.sgy  = fminf(fmaxf(r.xy2y / 127.0f * 2.0f - 1.0f, -1.1f), 1.1f);
    r.bound = b;
    return r;
}

// ---------------- Kernel A: xys for last camera, projection via V_WMMA_F32_16X16X4_F32 ----------------
// Each wave handles 16 grid points. A = R (3x3 padded to 16x4), B = (p - T) (3xN padded to 4x16).
// D rows 0..2, lanes 0..15 => xcam x/y/z for point = lane.
__global__ __launch_bounds__(256) void xys_wmma_kernel(
    const float* __restrict__ R, const float* __restrict__ T,
    const float* __restrict__ F, const float* __restrict__ C,
    const float* __restrict__ K, const float* __restrict__ P,
    const float* __restrict__ CTR, const float* __restrict__ SCL,
    float* __restrict__ out_xys)
{
    const int cam  = N_CAMS - 1;
    int lane = threadIdx.x & 31;
    int wave = threadIdx.x >> 5;
    int m    = lane & 15;
    int point = (blockIdx.x * 8 + wave) * 16 + m;   // 1000 blocks * 8 waves * 16 = 128000 exactly

    float gx, gy, gz;
    grid_point(point, gx, gy, gz);
    float dx = gx - T[cam * 3 + 0];
    float dy = gy - T[cam * 3 + 1];
    float dz = gz - T[cam * 3 + 2];

    const float* Rc = R + cam * 9;
    bool lo = lane < 16;

    // A 16x4 f32 layout: lanes 0-15 -> M=lane, VGPR0=K0, VGPR1=K1; lanes 16-31 -> M=lane-16, VGPR0=K2, VGPR1=K3
    v2f a;
    a[0] = (m < 3) ? (lo ? Rc[m * 3 + 0] : Rc[m * 3 + 2]) : 0.0f;
    a[1] = (m < 3 && lo) ? Rc[m * 3 + 1] : 0.0f;
    // B 4x16 f32 (mirrored): lanes 0-15 -> N=lane, VGPR0=K0, VGPR1=K1; lanes 16-31 -> N=lane-16, VGPR0=K2, VGPR1=K3(pad)
    v2f b;
    b[0] = lo ? dx : dz;
    b[1] = lo ? dy : 0.0f;

    v8f acc = {};
    acc = __builtin_amdgcn_wmma_f32_16x16x4_f32(
        /*neg_a=*/false, a, /*neg_b=*/false, b,
        /*c_mod=*/(short)0, acc, /*reuse_a=*/false, /*reuse_b=*/false);

    // lanes 0-15: acc[r] = xcam[r] for point = lane
    float xcx = acc[0], xcy = acc[1], xcz = acc[2];

    Proj pr = cam_transform(xcx, xcy, xcz,
        K[cam * 3 + 0], K[cam * 3 + 1], K[cam * 3 + 2],
        P[cam * 2 + 0], P[cam * 2 + 1],
        F[cam * 2 + 0], F[cam * 2 + 1],
        C[cam * 2 + 0], C[cam * 2 + 1],
        CTR[cam * 2 + 0], CTR[cam * 2 + 1], SCL[cam * 2 + 0]);

    if (lo) {
        out_xys[point * 2 + 0] = pr.xy2x;
        out_xys[point * 2 + 1] = pr.xy2y;
    }
}

// ---------------- Kernel B: fused per-voxel projection + bilinear sampling + accumulation ----------------
__global__ __launch_bounds__(256) void fuse_kernel(
    const float* __restrict__ heatmaps,   // [5,1,15,128,240]
    const float* __restrict__ cloud,      // [5,1,1,3,128,240]
    const float* __restrict__ R, const float* __restrict__ T,
    const float* __restrict__ F, const float* __restrict__ C,
    const float* __restrict__ K, const float* __restrict__ P,
    const float* __restrict__ CTR, const float* __restrict__ SCL,
    float* __restrict__ cubes)            // [15, NBINS]
{
    int n = blockIdx.x * blockDim.x + threadIdx.x;
    if (n >= NBINS) return;

    float gx, gy, gz;
    grid_point(n, gx, gy, gz);

    float num[N_JOINTS];
    #pragma unroll
    for (int j = 0; j < N_JOINTS; ++j) num[j] = 0.0f;
    float denom = 1e-6f;

    for (int cam = 0; cam < N_CAMS; ++cam) {
        const float* Rc = R + cam * 9;
        float dx = gx - T[cam * 3 + 0];
        float dy = gy - T[cam * 3 + 1];
        float dz = gz - T[cam * 3 + 2];
        float xcx = Rc[0] * dx + Rc[1] * dy + Rc[2] * dz;
        float xcy = Rc[3] * dx + Rc[4] * dy + Rc[5] * dz;
        float xcz = Rc[6] * dx + Rc[7] * dy + Rc[8] * dz;

        Proj pr = cam_transform(xcx, xcy, xcz,
            K[cam * 3 + 0], K[cam * 3 + 1], K[cam * 3 + 2],
            P[cam * 2 + 0], P[cam * 2 + 1],
            F[cam * 2 + 0], F[cam * 2 + 1],
            C[cam * 2 + 0], C[cam * 2 + 1],
            CTR[cam * 2 + 0], CTR[cam * 2 + 1], SCL[cam * 2 + 0]);

        denom += pr.bound;
        if (pr.bound == 0.0f) continue;   // contrib multiplied by 0 in reference; sg values finite -> equivalent

        // bilinear setup (align_corners=True, zero padding), shared by cloud + all 15 joints
        float xf = (pr.sgx + 1.0f) * ((HMW - 1) * 0.5f);
        float yf = (pr.sgy + 1.0f) * ((HMH - 1) * 0.5f);
        float x0f = floorf(xf), y0f = floorf(yf);
        float wx = xf - x0f, wy = yf - y0f;
        int ix0 = (int)x0f, iy0 = (int)y0f;
        int ix1 = ix0 + 1,  iy1 = iy0 + 1;
        bool vx0 = (ix0 >= 0) && (ix0 <= HMW - 1);
        bool vx1 = (ix1 >= 0) && (ix1 <= HMW - 1);
        bool vy0 = (iy0 >= 0) && (iy0 <= HMH - 1);
        bool vy1 = (iy1 >= 0) && (iy1 <= HMH - 1);
        int cx0 = min(max(ix0, 0), HMW - 1);
        int cx1 = min(max(ix1, 0), HMW - 1);
        int cy0 = min(max(iy0, 0), HMH - 1);
        int cy1 = min(max(iy1, 0), HMH - 1);
        float w00 = (vx0 && vy0) ? (1.0f - wx) * (1.0f - wy) : 0.0f;
        float w10 = (vx1 && vy0) ? wx * (1.0f - wy)          : 0.0f;
        float w01 = (vx0 && vy1) ? (1.0f - wx) * wy          : 0.0f;
        float w11 = (vx1 && vy1) ? wx * wy                   : 0.0f;
        int o00 = cy0 * HMW + cx0;
        int o10 = cy0 * HMW + cx1;
        int o01 = cy1 * HMW + cx0;
        int o11 = cy1 * HMW + cx1;

        // sample world-coord cloud -> gaussian weight
        const float* cl = cloud + (size_t)cam * 3 * HW;
        float s0 = w00 * cl[o00]          + w10 * cl[o10]          + w01 * cl[o01]          + w11 * cl[o11];
        float s1 = w00 * cl[HW + o00]     + w10 * cl[HW + o10]     + w01 * cl[HW + o01]     + w11 * cl[HW + o11];
        float s2 = w00 * cl[2*HW + o00]   + w10 * cl[2*HW + o10]   + w01 * cl[2*HW + o01]   + w11 * cl[2*HW + o11];
        float e0 = gx - s0, e1 = gy - s1, e2 = gz - s2;
        float d2 = e0 * e0 + e1 * e1 + e2 * e2;
        float wgt = 2.0f * __expf(0.0f) * 0.0f;  // placeholder avoided below
        wgt = 2.0f * expf(-d2 * (1.0f / (2.0f * 200.0f * 200.0f)));   // 2 * sampled * weight

        // sample all joints, reusing corner offsets/weights
        const float* hmc = heatmaps + (size_t)cam * N_JOINTS * HW;
        #pragma unroll
        for (int j = 0; j < N_JOINTS; ++j) {
            const float* h = hmc + j * HW;
            float s = w00 * h[o00] + w10 * h[o10] + w01 * h[o01] + w11 * h[o11];
            num[j] += s * wgt;
        }
    }

    #pragma unroll
    for (int j = 0; j < N_JOINTS; ++j) {
        float v = num[j] / denom;
        if (v != v) v = 0.0f;                 // isnan -> 0
        v = fminf(fmaxf(v, 0.0f), 1.0f);
        cubes[(size_t)j * NBINS + n] = v;
    }
}

// ---------------- Kernel C: grids = broadcast grid to [15, NBINS, 3] ----------------
__global__ __launch_bounds__(256) void grids_kernel(float* __restrict__ out)
{
    int idx = blockIdx.x * blockDim.x + threadIdx.x;
    const int total = N_JOINTS * NBINS * 3;   // 5,760,000
    if (idx >= total) return;
    int c = idx % 3;
    int t = idx / 3;
    int n = t % NBINS;
    float gx, gy, gz;
    grid_point(n, gx, gy, gz);
    out[idx] = (c == 0) ? gx : ((c == 1) ? gy : gz);
}

extern "C" void kernel_launch(void* const* d_in, const int* in_sizes, int n_in,
                              void* d_out, int out_size, void* d_ws, size_t ws_size,
                              hipStream_t stream) {
    (void)in_sizes; (void)n_in; (void)out_size; (void)d_ws; (void)ws_size;
    const float* heatmaps = (const float*)d_in[0];
    const float* cloud    = (const float*)d_in[1];
    const float* R        = (const float*)d_in[2];
    const float* T        = (const float*)d_in[3];
    const float* F        = (const float*)d_in[4];
    const float* C        = (const float*)d_in[5];
    const float* K        = (const float*)d_in[6];
    const float* P        = (const float*)d_in[7];
    const float* CTR      = (const float*)d_in[8];
    const float* SCL      = (const float*)d_in[9];

    float* cubes = (float*)d_out;                              // 15*128000
    float* grids = cubes + (size_t)N_JOINTS * NBINS;           // 15*128000*3
    float* xys   = grids + (size_t)N_JOINTS * NBINS * 3;       // 128000*2

    // Kernel B: 128000 voxels, one per lane
    fuse_kernel<<<(NBINS + 255) / 256, 256, 0, stream>>>(
        heatmaps, cloud, R, T, F, C, K, P, CTR, SCL, cubes);

    // Kernel A: 1000 blocks * 8 waves * 16 points = 128000 (WMMA projection, last camera)
    xys_wmma_kernel<<<1000, 256, 0, stream>>>(R, T, F, C, K, P, CTR, SCL, xys);

    // Kernel C: broadcast grid
    grids_kernel<<<(N_JOINTS * NBINS * 3 + 255) / 256, 256, 0, stream>>>(grids);
}